// RandomBlockMasker_36713380446765
// MI455X (gfx1250) — compile-verified
//
#include <hip/hip_runtime.h>

// [B, C, H, W] = [32, 3, 512, 512] fp32, K = 4 blocks of 64x64 per batch.
// Pure streaming kernel: ~201 MB of traffic, HBM-bound (~8.6 us floor @ 23.3 TB/s).
// WMMA is intentionally NOT used: there is no matrix contraction in this op,
// and the chip rewards b128 vector memory + NT cache hints here instead
// (working set 201 MB > 192 MB L2, each byte touched exactly once).

typedef float v4f __attribute__((ext_vector_type(4)));

static constexpr int Bn  = 32;
static constexpr int Cn  = 3;
static constexpr int Hn  = 512;
static constexpr int Wn  = 512;
static constexpr int Kn  = 4;
static constexpr int BLK = 64;

__global__ __launch_bounds__(128)
void RandomBlockMasker_36713380446765_kernel(const float* __restrict__ x,
                                             const int*   __restrict__ tops,
                                             const int*   __restrict__ lefts,
                                             float*       __restrict__ out) {
    // One workgroup per (b, h) row; 128 threads x 4 floats cover W = 512.
    const int bh = blockIdx.x;
    const int b  = bh >> 9;          // H = 512 = 2^9
    const int h  = bh & (Hn - 1);
    const int w0 = threadIdx.x << 2; // 4 consecutive w per thread (16B aligned)

    // Mask for the 4 pixels this thread owns. tops/lefts indices are
    // block-uniform (depend only on b) -> scalar loads (s_load_b32).
    bool m0 = false, m1 = false, m2 = false, m3 = false;
#pragma unroll
    for (int k = 0; k < Kn; ++k) {
        const int t = tops[b * Kn + k];
        const int l = lefts[b * Kn + k];
        const bool inh = (h >= t) & (h < t + BLK);
        m0 |= inh & (w0 + 0 >= l) & (w0 + 0 < l + BLK);
        m1 |= inh & (w0 + 1 >= l) & (w0 + 1 < l + BLK);
        m2 |= inh & (w0 + 2 >= l) & (w0 + 2 < l + BLK);
        m3 |= inh & (w0 + 3 >= l) & (w0 + 3 < l + BLK);
    }
    // Keep factor: clamp(x,0,1) >= 0, so multiply-by-0/1 implements the select.
    const float k0 = m0 ? 0.0f : 1.0f;
    const float k1 = m1 ? 0.0f : 1.0f;
    const float k2 = m2 ? 0.0f : 1.0f;
    const float k3 = m3 ? 0.0f : 1.0f;

    // Mask is channel-independent: reuse it across C = 3 channels.
#pragma unroll
    for (int c = 0; c < Cn; ++c) {
        const size_t idx = ((((size_t)b * Cn + c) * Hn + h) * Wn) + (size_t)w0;
        // Non-temporal b128 load: stream, don't pollute L2 (read-once data).
        v4f v = __builtin_nontemporal_load((const v4f*)(x + idx));
        v4f r;
        r.x = fminf(fmaxf(v.x, 0.0f), 1.0f) * k0;
        r.y = fminf(fmaxf(v.y, 0.0f), 1.0f) * k1;
        r.z = fminf(fmaxf(v.z, 0.0f), 1.0f) * k2;
        r.w = fminf(fmaxf(v.w, 0.0f), 1.0f) * k3;
        // Non-temporal b128 store: write-once output stream.
        __builtin_nontemporal_store(r, (v4f*)(out + idx));
    }
}

extern "C" void kernel_launch(void* const* d_in, const int* in_sizes, int n_in,
                              void* d_out, int out_size, void* d_ws, size_t ws_size,
                              hipStream_t stream) {
    const float* x     = (const float*)d_in[0]; // [32,3,512,512] fp32
    const int*   tops  = (const int*)d_in[1];   // [32,4] int32
    const int*   lefts = (const int*)d_in[2];   // [32,4] int32
    float*       out   = (float*)d_out;         // [32,3,512,512] fp32

    const int grid  = Bn * Hn;                  // 16384 workgroups, one per (b,h)
    const int block = Wn / 4;                   // 128 threads = 4 wave32

    RandomBlockMasker_36713380446765_kernel<<<grid, block, 0, stream>>>(x, tops, lefts, out);
}